// GCNBlock_58308476010758
// MI455X (gfx1250) — compile-verified
//
#include <hip/hip_runtime.h>

// ---------------- problem constants (match reference) ----------------
#define NB   256                  // graphs
#define NN   360                  // nodes per graph
#define KK   180                  // TopK keep == NN/2 (also half-tile size)
#define DD   64                   // feature dim
#define NTOT (NB * NN)            // 92160 nodes
#define EPG  (NN * 32)            // 11520 edges per graph (edges are graph-grouped)
#define NE   (NB * EPG)           // 2949120 edges
#define BN_EPS 1e-5f

typedef float v2f __attribute__((ext_vector_type(2)));
typedef float v8f __attribute__((ext_vector_type(8)));

// ---------------- K1: zero BN stats + 1/||p|| ----------------
__global__ void k_pnorm_init(const float* __restrict__ p, float* __restrict__ sums) {
    __shared__ float red[64];
    int t = threadIdx.x;
    sums[t] = 0.0f;            // channel sums
    sums[64 + t] = 0.0f;       // channel sumsq
    red[t] = p[t] * p[t];
    __syncthreads();
    for (int off = 32; off > 0; off >>= 1) {
        if (t < off) red[t] += red[t + off];
        __syncthreads();
    }
    if (t == 0) sums[128] = rsqrtf(red[0]);
}

// ---------------- K2: per-graph degree in LDS -> dinv (fused rsqrt) ----------------
__global__ void k_deg_lds(const int* __restrict__ col, const float* __restrict__ w,
                          float* __restrict__ dinv) {
    __shared__ float sh_deg[NN];
    int g = blockIdx.x, tid = threadIdx.x;
    for (int j = tid; j < NN; j += 256) sh_deg[j] = 0.0f;
    __syncthreads();
    int ebase = g * EPG;
    for (int el = tid; el < EPG; el += 256) {
        int e = ebase + el;
        atomicAdd(&sh_deg[col[e] - g * NN], w[e]);   // ds_add_f32
    }
    __syncthreads();
    for (int j = tid; j < NN; j += 256) {
        float d = 1.0f + sh_deg[j];                  // + self-loop weight 1
        dinv[g * NN + j] = (d > 0.0f) ? rsqrtf(d) : 0.0f;
    }
}

// ---------------- K3: h = x @ W via V_WMMA_F32_16X16X4_F32 ----------------
// W kept transposed in LDS (sWt[n][k], stride 68 => banks (4n+ka)%64, conflict-free),
// so each B fragment is one contiguous 8B LDS load straight into a VGPR pair.
__global__ void k_gemm(const float* __restrict__ x, const float* __restrict__ W,
                       float* __restrict__ h) {
    __shared__ float sWt[64 * 68];
    int tid = threadIdx.x;
    for (int i = tid; i < 64 * 64; i += 256) {
        int r = i >> 6, c = i & 63;
        sWt[c * 68 + r] = W[i];                      // transpose: sWt[n*68 + k] = W[k][n]
    }
    __syncthreads();

    int wave = tid >> 5, lane = tid & 31;
    int tile = blockIdx.x * 8 + wave;                // 720 blocks * 8 waves = 5760 tiles
    int row0 = tile * 16;
    int m = lane & 15, hi = lane >> 4;

    v8f acc[4] = {v8f{}, v8f{}, v8f{}, v8f{}};
    const float* xr = x + (size_t)(row0 + m) * DD;

    for (int k4 = 0; k4 < DD; k4 += 4) {
        int ka = k4 + 2 * hi;                        // A 16x4: lanes 0-15 K={0,1}, 16-31 K={2,3}
        v2f a = *(const v2f*)(xr + ka);
#pragma unroll
        for (int nt = 0; nt < 4; ++nt) {
            int n = nt * 16 + m;                     // B 4x16 mirrors A layout
            v2f b = *(const v2f*)&sWt[n * 68 + ka];  // {W[ka][n], W[ka+1][n]} contiguous
            acc[nt] = __builtin_amdgcn_wmma_f32_16x16x4_f32(
                false, a, false, b, (short)0, acc[nt], false, false);
        }
    }
    // D 16x16 f32 layout: VGPR v -> row v + 8*hi, col = lane&15
#pragma unroll
    for (int v = 0; v < 8; ++v) {
        float* hr = h + (size_t)(row0 + v + 8 * hi) * DD + m;
        hr[0]  = acc[0][v];
        hr[16] = acc[1][v];
        hr[32] = acc[2][v];
        hr[48] = acc[3][v];
    }
}

// ---------------- K4: GCN aggregate per (graph, half) tile in LDS ----------------
// Replaces 189M global fp32 atomics with ds_add_f32 into a block-private tile,
// then fuses self-loop + bias + ReLU into the single non-atomic writeout.
__global__ void k_conv(const int* __restrict__ row, const int* __restrict__ col,
                       const float* __restrict__ wgt, const float* __restrict__ dinv,
                       const float* __restrict__ h, const float* __restrict__ bvec,
                       float* __restrict__ out) {
    __shared__ float sh_out[KK * DD];                // 46080 B accumulation tile
    __shared__ float sh_dinv[NN];
    int g = blockIdx.x >> 1, half = blockIdx.x & 1;
    int tid = threadIdx.x, lane = tid & 31, wv = tid >> 5;

    for (int i = tid; i < KK * DD; i += 256) sh_out[i] = 0.0f;
    for (int j = tid; j < NN; j += 256) sh_dinv[j] = dinv[g * NN + j];
    __syncthreads();

    int ebase = g * EPG;
    int d0 = lane * 2;
    for (int el = wv; el < EPG; el += 8) {           // one edge per wave per step
        int e = ebase + el;
        __builtin_prefetch(&col[e + 2048], 0, 1);    // global_prefetch_b8 (edge stream)
        int c = col[e];
        int cl = c - g * NN - half * KK;
        if (cl >= 0 && cl < KK) {
            int r = row[e];
            float nrm = sh_dinv[r - g * NN] * wgt[e] * sh_dinv[c - g * NN];
            v2f hv = *(const v2f*)(h + (size_t)r * DD + d0);   // coalesced b64
            atomicAdd(&sh_out[cl * DD + d0],     nrm * hv.x);  // ds_add_f32, banks {2L,2L+1}
            atomicAdd(&sh_out[cl * DD + d0 + 1], nrm * hv.y);
        }
    }
    __syncthreads();

    int nbase = g * NN + half * KK;
    for (int i = tid; i < KK * DD; i += 256) {
        int nl = i >> 6, d = i & 63;
        int node = nbase + nl;
        float di = sh_dinv[half * KK + nl];
        float v = sh_out[i] + di * di * h[(size_t)node * DD + d] + bvec[d];
        out[(size_t)node * DD + d] = fmaxf(v, 0.0f); // fused ReLU
    }
}

// ---------------- K5: per-channel sum / sumsq (read-only pass) ----------------
__global__ void k_stats(const float* __restrict__ out, float* __restrict__ sums) {
    const int RSTRIDE = 1536;                        // 384 blocks * 256 / 64 channels
    int t = blockIdx.x * 256 + threadIdx.x;
    int d = t & 63, r0 = t >> 6;
    float s = 0.0f, s2 = 0.0f;
    for (int r = r0; r < NTOT; r += RSTRIDE) {
        float v = out[(size_t)r * DD + d];
        s += v; s2 += v * v;
    }
    atomicAdd(&sums[d], s);
    atomicAdd(&sums[64 + d], s2);
}

// ---------------- K6: BN apply in place + score (wave32 shfl reduction) ----------------
__global__ void k_bn_score(float* __restrict__ out, const float* __restrict__ sums,
                           const float* __restrict__ gamma, const float* __restrict__ beta,
                           const float* __restrict__ p, float* __restrict__ score) {
    int tid = threadIdx.x;
    int node = blockIdx.x * 8 + (tid >> 5);
    int lane = tid & 31;
    int d0 = lane * 2;
    const float inv_n = 1.0f / (float)NTOT;

    float mu0 = sums[d0] * inv_n,     mu1 = sums[d0 + 1] * inv_n;
    float vr0 = sums[64 + d0] * inv_n - mu0 * mu0;
    float vr1 = sums[64 + d0 + 1] * inv_n - mu1 * mu1;
    float sc0 = rsqrtf(vr0 + BN_EPS) * gamma[d0];
    float sc1 = rsqrtf(vr1 + BN_EPS) * gamma[d0 + 1];

    size_t base = (size_t)node * DD + d0;
    float v0 = (out[base]     - mu0) * sc0 + beta[d0];
    float v1 = (out[base + 1] - mu1) * sc1 + beta[d0 + 1];
    out[base]     = v0;
    out[base + 1] = v1;

    float partial = v0 * p[d0] + v1 * p[d0 + 1];
#pragma unroll
    for (int off = 16; off > 0; off >>= 1)
        partial += __shfl_xor(partial, off, 32);
    if (lane == 0) score[node] = partial * sums[128];
}

// ---------------- K7: per-graph bitonic TopK (1 block per graph) ----------------
__global__ void k_topk(const float* __restrict__ score, int* __restrict__ perm,
                       float* __restrict__ gate, int* __restrict__ new_id,
                       float* __restrict__ batch_out) {
    __shared__ float ss[512];
    __shared__ int   si[512];
    int g = blockIdx.x, tid = threadIdx.x;
    for (int j = tid; j < 512; j += 256) {
        ss[j] = (j < NN) ? score[g * NN + j] : -1e30f;
        si[j] = j;
    }
    __syncthreads();
    // descending by score, ascending index on ties (matches lax.top_k)
    for (int k = 2; k <= 512; k <<= 1) {
        for (int jj = k >> 1; jj > 0; jj >>= 1) {
            for (int base = 0; base < 512; base += 256) {
                int i = base + tid;
                int ixj = i ^ jj;
                if (ixj > i) {
                    float s1 = ss[i], s2 = ss[ixj];
                    int   i1 = si[i], i2 = si[ixj];
                    bool before = (s1 > s2) || (s1 == s2 && i1 < i2);
                    bool dirDesc = ((i & k) == 0);
                    if (dirDesc ? !before : before) {
                        ss[i] = s2; ss[ixj] = s1;
                        si[i] = i2; si[ixj] = i1;
                    }
                }
            }
            __syncthreads();
        }
    }
    for (int j = tid; j < NN; j += 256) new_id[g * NN + j] = -1;
    __syncthreads();
    if (tid < KK) {
        int node = si[tid];
        perm[g * KK + tid]      = g * NN + node;
        gate[g * KK + tid]      = tanhf(ss[tid]);
        new_id[g * NN + node]   = g * KK + tid;
        batch_out[g * KK + tid] = (float)g;
    }
}

// ---------------- K8: gather x_pool = out[perm] * gate ----------------
__global__ void k_gather(const float* __restrict__ out, const int* __restrict__ perm,
                         const float* __restrict__ gate, float* __restrict__ xp) {
    size_t t = (size_t)blockIdx.x * 256 + threadIdx.x;
    if (t < (size_t)NB * KK * DD) {
        int r = (int)(t >> 6), d = (int)(t & 63);
        xp[t] = out[(size_t)perm[r] * DD + d] * gate[r];
    }
}

// ---------------- K9: edge filter / remap ----------------
__global__ void k_edges(const int* __restrict__ row, const int* __restrict__ col,
                        const float* __restrict__ w, const int* __restrict__ new_id,
                        float* __restrict__ ei, float* __restrict__ ea,
                        float* __restrict__ mk) {
    int e = blockIdx.x * 256 + threadIdx.x;
    if (e < NE) {
        int nr = new_id[row[e]], nc = new_id[col[e]];
        bool keep = (nr >= 0) && (nc >= 0);
        ei[e]      = keep ? (float)nr : -1.0f;
        ei[NE + e] = keep ? (float)nc : -1.0f;
        ea[e]      = keep ? w[e] : 0.0f;
        mk[e]      = keep ? 1.0f : 0.0f;
    }
}

// ---------------- launch ----------------
extern "C" void kernel_launch(void* const* d_in, const int* in_sizes, int n_in,
                              void* d_out, int out_size, void* d_ws, size_t ws_size,
                              hipStream_t stream) {
    const float* x  = (const float*)d_in[0];
    const int*   ei = (const int*)  d_in[1];    // [2, NE]
    const float* w  = (const float*)d_in[2];
    const float* W  = (const float*)d_in[4];
    const float* b  = (const float*)d_in[5];
    const float* gm = (const float*)d_in[6];
    const float* bt = (const float*)d_in[7];
    const float* p  = (const float*)d_in[8];
    const int* row = ei;
    const int* col = ei + NE;

    // workspace layout (floats)
    float* ws    = (float*)d_ws;
    float* h     = ws;                         // 5,898,240
    float* out   = h + (size_t)NTOT * DD;      // 5,898,240
    float* dinv  = out + (size_t)NTOT * DD;    // 92,160
    float* score = dinv + NTOT;                // 92,160
    float* sums  = score + NTOT;               // 192
    int*   perm  = (int*)(sums + 192);         // 46,080
    float* gate  = (float*)(perm + NB * KK);   // 46,080
    int*   nid   = (int*)(gate + NB * KK);     // 92,160

    // output layout (floats, tuple order)
    float* o    = (float*)d_out;
    float* o_xp = o;                           // B*K*D
    float* o_ei = o_xp + (size_t)NB * KK * DD; // 2*NE
    float* o_ea = o_ei + (size_t)2 * NE;       // NE
    float* o_bp = o_ea + (size_t)NE;           // B*K
    float* o_mk = o_bp + (size_t)NB * KK;      // NE

    k_pnorm_init<<<1,                 64, 0, stream>>>(p, sums);
    k_deg_lds   <<<NB,               256, 0, stream>>>(col, w, dinv);
    k_gemm      <<<NTOT / 16 / 8,    256, 0, stream>>>(x, W, h);
    k_conv      <<<NB * 2,           256, 0, stream>>>(row, col, w, dinv, h, b, out);
    k_stats     <<<384,              256, 0, stream>>>(out, sums);
    k_bn_score  <<<NTOT / 8,         256, 0, stream>>>(out, sums, gm, bt, p, score);
    k_topk      <<<NB,               256, 0, stream>>>(score, perm, gate, nid, o_bp);
    k_gather    <<<NB * KK * DD / 256, 256, 0, stream>>>(out, perm, gate, o_xp);
    k_edges     <<<(NE + 255) / 256, 256, 0, stream>>>(row, col, w, nid, o_ei, o_ea, o_mk);
}